// MessagePassing_76768245449703
// MI455X (gfx1250) — compile-verified
//
#include <hip/hip_runtime.h>

// Problem constants (from reference): B=8, N=1024, D=128, E=8
#define PB 8
#define PN 1024
#define PD 128
#define PE 8

typedef float v2f __attribute__((ext_vector_type(2)));
typedef float v4f __attribute__((ext_vector_type(4)));
typedef float v8f __attribute__((ext_vector_type(8)));

// ---------------------------------------------------------------------------
// Kernel A: coef[b,s,e] = sum_o adj[b,s,o] * edge[b,s,o,e]
// One 256-thread block per (b,s). 288 MB streamed once -> NT loads.
// ---------------------------------------------------------------------------
__global__ __launch_bounds__(256) void coef_kernel(
    const float* __restrict__ adj,    // [B,N,N]
    const float* __restrict__ edge,   // [B,N,N,E]
    float* __restrict__ coef)         // [B,N,E]  (workspace)
{
    const int bs = blockIdx.x;                 // b*N + s
    const int t  = threadIdx.x;                // 0..255
    const float* __restrict__ adj_row  = adj  + (size_t)bs * PN;
    const float* __restrict__ edge_row = edge + (size_t)bs * PN * PE;

    v4f acc0 = {0.f, 0.f, 0.f, 0.f};
    v4f acc1 = {0.f, 0.f, 0.f, 0.f};

    #pragma unroll
    for (int it = 0; it < PN / 256; ++it) {
        const int o = t + it * 256;
        const float a = __builtin_nontemporal_load(adj_row + o);
        const v4f* ep = (const v4f*)(edge_row + (size_t)o * PE);
        const v4f e0 = __builtin_nontemporal_load(ep + 0);  // e = 0..3
        const v4f e1 = __builtin_nontemporal_load(ep + 1);  // e = 4..7
        acc0 += a * e0;
        acc1 += a * e1;
    }

    // wave32 butterfly reduction (warpSize == 32 on gfx1250)
    #pragma unroll
    for (int off = 16; off > 0; off >>= 1) {
        acc0.x += __shfl_xor(acc0.x, off, 32);
        acc0.y += __shfl_xor(acc0.y, off, 32);
        acc0.z += __shfl_xor(acc0.z, off, 32);
        acc0.w += __shfl_xor(acc0.w, off, 32);
        acc1.x += __shfl_xor(acc1.x, off, 32);
        acc1.y += __shfl_xor(acc1.y, off, 32);
        acc1.z += __shfl_xor(acc1.z, off, 32);
        acc1.w += __shfl_xor(acc1.w, off, 32);
    }

    __shared__ float red[8][8];                // 8 waves x 8 e-values
    const int lane = t & 31;
    const int wv   = t >> 5;
    if (lane == 0) {
        red[wv][0] = acc0.x; red[wv][1] = acc0.y;
        red[wv][2] = acc0.z; red[wv][3] = acc0.w;
        red[wv][4] = acc1.x; red[wv][5] = acc1.y;
        red[wv][6] = acc1.z; red[wv][7] = acc1.w;
    }
    __syncthreads();
    if (t < PE) {
        float s = 0.f;
        #pragma unroll
        for (int w = 0; w < 8; ++w) s += red[w][t];
        coef[(size_t)bs * PE + t] = s;
    }
}

// ---------------------------------------------------------------------------
// Kernel B (fused): out[b,s,i] = sum_e coef[b,s,e] * (NS[b,s,:] . W[e,i,:])
// One block per 16-row s-tile; 8 waves, wave w owns output cols [16w,16w+16).
// f32 WMMA 16x16x4, K=128 per e, accumulate across e with per-row coef scale.
// ---------------------------------------------------------------------------
#define NS_STRIDE 132   // 128 + 4 pad: keeps v4 LDS stores 16B-aligned and
                        // makes the K-major 16-lane gather bank-conflict-free

__global__ __launch_bounds__(256) void fused_mp_kernel(
    const float* __restrict__ ns,     // [B,N,D]
    const float* __restrict__ W,      // [E,D,D]  W[e,i,j]
    const float* __restrict__ coef,   // [B,N,E]
    float* __restrict__ out)          // [B,N,D]
{
    __shared__ float ns_t[16 * NS_STRIDE];     // 16 x 128 NS tile (padded)
    __shared__ float coef_t[16 * PE];          // 16 rows x 8 e

    const int blk  = blockIdx.x;               // 0 .. B*(N/16)-1 = 511
    const int b    = blk >> 6;                 // N/16 = 64 tiles per batch
    const int s0   = (blk & 63) << 4;
    const int t    = threadIdx.x;
    const int lane = t & 31;
    const int wv   = t >> 5;                   // 0..7
    const int half = lane >> 4;                // 0 or 1 (lane 0-15 / 16-31)
    const int m    = lane & 15;

    // Stage NS tile: 16 rows x 128 cols = 512 v4f, 2 per thread
    {
        const float* src = ns + ((size_t)(b * PN + s0)) * PD;
        #pragma unroll
        for (int i = t; i < 512; i += 256) {
            const int r  = i >> 5;             // 32 v4f per row
            const int c4 = (i & 31) << 2;
            v4f v = *(const v4f*)(src + r * PD + c4);
            *(v4f*)&ns_t[r * NS_STRIDE + c4] = v;
        }
        if (t < 16 * PE)
            coef_t[t] = coef[(size_t)(b * PN + s0) * PE + t];
    }
    __syncthreads();

    // A fragments (16x4 f32 layout: lanes 0-15 K={0,1}, lanes 16-31 K={2,3})
    v2f a[32];
    {
        const float* arow = &ns_t[m * NS_STRIDE + half * 2];
        #pragma unroll
        for (int kb = 0; kb < 32; ++kb)
            a[kb] = *(const v2f*)(arow + kb * 4);   // ds_load_b64
    }

    const int icol = (wv << 4) + m;            // output feature column
    v8f oacc = {0.f, 0.f, 0.f, 0.f, 0.f, 0.f, 0.f, 0.f};

    for (int e = 0; e < PE; ++e) {
        // B fragment source: B[K=j, N=i] = W[e, i, j]; lane holds row i=icol,
        // regs K = 4kb + 2*half + {0,1}  ->  8-byte loads, hot in WGP$/L2.
        const float* wp = W + ((size_t)e << 14) + (size_t)icol * PD + half * 2;
        v8f c = {0.f, 0.f, 0.f, 0.f, 0.f, 0.f, 0.f, 0.f};
        #pragma unroll
        for (int kb = 0; kb < 32; ++kb) {
            const v2f bfrag = *(const v2f*)(wp + kb * 4);
            c = __builtin_amdgcn_wmma_f32_16x16x4_f32(
                    /*neg_a=*/false, a[kb],
                    /*neg_b=*/false, bfrag,
                    /*c_mod=*/(short)0, c,
                    /*reuse_a=*/false, /*reuse_b=*/false);
        }
        // C/D layout: VGPR r -> row M = r + 8*half, col N = m
        #pragma unroll
        for (int r = 0; r < 8; ++r) {
            const int row = r + (half << 3);
            oacc[r] += coef_t[row * PE + e] * c[r];
        }
    }

    // Store: out[b, s0+row, icol]
    float* op = out + ((size_t)(b * PN + s0)) * PD + icol;
    #pragma unroll
    for (int r = 0; r < 8; ++r) {
        const int row = r + (half << 3);
        op[(size_t)row * PD] = oacc[r];
    }
}

// ---------------------------------------------------------------------------
extern "C" void kernel_launch(void* const* d_in, const int* in_sizes, int n_in,
                              void* d_out, int out_size, void* d_ws, size_t ws_size,
                              hipStream_t stream) {
    const float* node_state = (const float*)d_in[0];   // [B,N,D]
    const float* edge_type  = (const float*)d_in[1];   // [B,N,N,E]
    const float* adj_mat    = (const float*)d_in[2];   // [B,N,N]
    const float* Wmat       = (const float*)d_in[3];   // [E,D,D]
    float* out  = (float*)d_out;                        // [B,N,D]
    float* coef = (float*)d_ws;                         // [B,N,E] = 256 KB

    coef_kernel<<<PB * PN, 256, 0, stream>>>(adj_mat, edge_type, coef);
    fused_mp_kernel<<<PB * (PN / 16), 256, 0, stream>>>(node_state, Wmat, coef, out);
}